// StockModel_90520730730564
// MI455X (gfx1250) — compile-verified
//
#include <hip/hip_runtime.h>
#include <math.h>

typedef float v2f __attribute__((ext_vector_type(2)));
typedef float v8f __attribute__((ext_vector_type(8)));

#define NEG_SLOPE 0.2f

// ---------- float -> order-preserving uint (for atomicMax on float) ----------
__device__ __forceinline__ unsigned f2ord(float f) {
  unsigned u = __float_as_uint(f);
  return (u & 0x80000000u) ? ~u : (u | 0x80000000u);
}
__device__ __forceinline__ float ord2f(unsigned k) {
  unsigned u = (k & 0x80000000u) ? (k & 0x7fffffffu) : ~k;
  return __uint_as_float(u);
}
__device__ __forceinline__ float lrelu(float t) { return t > 0.f ? t : t * NEG_SLOPE; }
__device__ __forceinline__ float elu(float t) { return t > 0.f ? t : (__expf(t) - 1.f); }

// ---------------------------- zero-init workspace ----------------------------
__global__ void zero_f4(float4* p, long n4) {
  long i = (long)blockIdx.x * blockDim.x + threadIdx.x;
  if (i < n4) p[i] = make_float4(0.f, 0.f, 0.f, 0.f);
}

// ------------------------------ WMMA f32 GEMM --------------------------------
// C[M x Nout] = A[M x K] @ W[K x Nout] + bias (row-major f32).
// 256-thread WG = 8 waves; WG covers 128 rows x 64 cols. The W panel (128x64
// per K-chunk) is staged into LDS with gfx1250 async global->LDS b128 loads
// (ASYNCcnt), then all 8 waves read B fragments from LDS. Padded LDS row
// stride of 72 floats makes the two half-wave B reads bank-conflict-free
// (2*72 mod 64 == 16). Per-wave guard is uniform -> EXEC all ones at WMMA.
template <int K>
__global__ __launch_bounds__(256) void gemm_wmma_lds(const float* __restrict__ A,
                                                     const float* __restrict__ W,
                                                     const float* __restrict__ bias,
                                                     float* __restrict__ C, int Nout,
                                                     int Mtiles) {
  constexpr int KC  = 128;  // K-chunk rows staged per iteration
  constexpr int PAD = 72;   // padded LDS row stride (floats)
  __shared__ float sW[KC * PAD];

  const int tid  = threadIdx.x;
  const int lane = tid & 31;
  const int half = lane >> 4;  // 0: K pair {0,1}, 1: K pair {2,3}
  const int l    = lane & 15;  // M (A) / N (B,C,D)
  const int mtile = blockIdx.x * 8 + (tid >> 5);
  const int n0    = blockIdx.y * 64;

  v8f acc[4];
#pragma unroll
  for (int j = 0; j < 4; ++j) acc[j] = (v8f){};

  const float* Arow = A + (size_t)(mtile * 16 + l) * K;

  for (int kc = 0; kc < K; kc += KC) {
    // ---- stage W[kc:kc+KC, n0:n0+64] into LDS (async, 8 x b128 per thread) --
#pragma unroll
    for (int i = 0; i < (KC * 64) / (256 * 4); ++i) {
      const int flat = (i * 256 + tid) * 4;
      const int r = flat >> 6;   // row within chunk
      const int n = flat & 63;   // col within panel
      const unsigned lds = (unsigned)(size_t)&sW[r * PAD + n];
      const unsigned long long ga =
          (unsigned long long)(size_t)(W + (size_t)(kc + r) * Nout + n0 + n);
      asm volatile("global_load_async_to_lds_b128 %0, %1, off"
                   :: "v"(lds), "v"(ga) : "memory");
    }
    asm volatile("s_wait_asynccnt 0x0" ::: "memory");
    __syncthreads();

    // ---- consume chunk ------------------------------------------------------
    if (mtile < Mtiles) {
      for (int k0 = 0; k0 < KC; k0 += 4) {
        const int kb = k0 + half * 2;
        v2f a;
        a.x = Arow[kc + kb];
        a.y = Arow[kc + kb + 1];
        const float* s0 = &sW[kb * PAD];
#pragma unroll
        for (int j = 0; j < 4; ++j) {
          v2f b;
          b.x = s0[j * 16 + l];
          b.y = s0[PAD + j * 16 + l];
          acc[j] = __builtin_amdgcn_wmma_f32_16x16x4_f32(false, a, false, b, (short)0,
                                                         acc[j], false, false);
        }
      }
    }
    __syncthreads();
  }

  if (mtile < Mtiles) {
    const int m0 = mtile * 16;
#pragma unroll
    for (int j = 0; j < 4; ++j) {
      const int n = n0 + j * 16 + l;
      const float bv = bias[n];
#pragma unroll
      for (int v = 0; v < 8; ++v) {
        const int m = m0 + v + half * 8;  // C/D layout: VGPR v -> M=v (+8 upper lanes)
        C[(size_t)m * Nout + n] = acc[j][v] + bv;
      }
    }
  }
}

// ------------------------- layer 1 edge kernels (H=4,C=64) -------------------
// One wave per edge; lane owns 8 channels (float4 x2); 8 lanes per head.
__global__ void edge_score_l1(const float* __restrict__ xl, const float* __restrict__ xr,
                              const int* __restrict__ src, const int* __restrict__ dst,
                              const float* __restrict__ att, float* __restrict__ score,
                              unsigned* __restrict__ maxb, int E) {
  const int wid = (int)((blockIdx.x * blockDim.x + threadIdx.x) >> 5);
  const int lane = threadIdx.x & 31;
  if (wid >= E) return;
  const int s = src[wid], d = dst[wid];
  const float4* pl = (const float4*)(xl + (size_t)s * 256);
  const float4* pr = (const float4*)(xr + (size_t)d * 256);
  const float4* pa = (const float4*)att;
  float sum = 0.f;
#pragma unroll
  for (int q = 0; q < 2; ++q) {
    const float4 a = pl[lane * 2 + q];
    const float4 b = pr[lane * 2 + q];
    const float4 w = pa[lane * 2 + q];
    sum += lrelu(a.x + b.x) * w.x;
    sum += lrelu(a.y + b.y) * w.y;
    sum += lrelu(a.z + b.z) * w.z;
    sum += lrelu(a.w + b.w) * w.w;
  }
  sum += __shfl_xor(sum, 1);
  sum += __shfl_xor(sum, 2);
  sum += __shfl_xor(sum, 4);  // lanes 8h..8h+7 now hold head-h sum
  if ((lane & 7) == 0) {
    const int h = lane >> 3;
    score[(size_t)wid * 4 + h] = sum;
    atomicMax(maxb + (size_t)d * 4 + h, f2ord(sum));
  }
}

template <int HH>
__global__ void edge_softmax(const int* __restrict__ dst, const unsigned* __restrict__ maxb,
                             float* __restrict__ sc /* in: score, out: exp */,
                             float* __restrict__ denom, int total) {
  const int i = blockIdx.x * blockDim.x + threadIdx.x;
  if (i >= total) return;
  const int e = i / HH, h = i % HH;
  const int d = dst[e];
  const float m = ord2f(maxb[(size_t)d * HH + h]);
  const float a = __expf(sc[i] - m);
  sc[i] = a;
  atomicAdd(denom + (size_t)d * HH + h, a);
}

__global__ void edge_aggr_l1(const float* __restrict__ xl, const int* __restrict__ src,
                             const int* __restrict__ dst, const float* __restrict__ expw,
                             const float* __restrict__ denom, float* __restrict__ agg, int E) {
  const int wid = (int)((blockIdx.x * blockDim.x + threadIdx.x) >> 5);
  const int lane = threadIdx.x & 31;
  if (wid >= E) return;
  const int s = src[wid], d = dst[wid];
  const int h = lane >> 3;
  const float alpha = expw[(size_t)wid * 4 + h] / denom[(size_t)d * 4 + h];
  const float4* pl = (const float4*)(xl + (size_t)s * 256);
  float* po = agg + (size_t)d * 256;
#pragma unroll
  for (int q = 0; q < 2; ++q) {
    const float4 a = pl[lane * 2 + q];
    const int c = (lane * 2 + q) * 4;
    atomicAdd(po + c + 0, a.x * alpha);
    atomicAdd(po + c + 1, a.y * alpha);
    atomicAdd(po + c + 2, a.z * alpha);
    atomicAdd(po + c + 3, a.w * alpha);
  }
}

__global__ void bias_elu(float* __restrict__ v, const float* __restrict__ bias, int chmask,
                         int total) {
  const int i = blockIdx.x * blockDim.x + threadIdx.x;
  if (i >= total) return;
  v[i] = elu(v[i] + bias[i & chmask]);
}

// ------------------------- layer 2 edge kernels (H=1,C=64) -------------------
__global__ void edge_score_l2(const float* __restrict__ xl, const float* __restrict__ xr,
                              const int* __restrict__ src, const int* __restrict__ dst,
                              const float* __restrict__ att, float* __restrict__ score,
                              unsigned* __restrict__ maxb, int E) {
  const int wid = (int)((blockIdx.x * blockDim.x + threadIdx.x) >> 5);
  const int lane = threadIdx.x & 31;
  if (wid >= E) return;
  const int s = src[wid], d = dst[wid];
  const float2 a = ((const float2*)(xl + (size_t)s * 64))[lane];
  const float2 b = ((const float2*)(xr + (size_t)d * 64))[lane];
  const float2 w = ((const float2*)att)[lane];
  float sum = lrelu(a.x + b.x) * w.x + lrelu(a.y + b.y) * w.y;
  sum += __shfl_xor(sum, 1);
  sum += __shfl_xor(sum, 2);
  sum += __shfl_xor(sum, 4);
  sum += __shfl_xor(sum, 8);
  sum += __shfl_xor(sum, 16);
  if (lane == 0) {
    score[wid] = sum;
    atomicMax(maxb + d, f2ord(sum));
  }
}

__global__ void edge_aggr_l2(const float* __restrict__ xl, const int* __restrict__ src,
                             const int* __restrict__ dst, const float* __restrict__ expw,
                             const float* __restrict__ denom, float* __restrict__ agg, int E) {
  const int wid = (int)((blockIdx.x * blockDim.x + threadIdx.x) >> 5);
  const int lane = threadIdx.x & 31;
  if (wid >= E) return;
  const int s = src[wid], d = dst[wid];
  const float alpha = expw[wid] / denom[d];
  const float2 a = ((const float2*)(xl + (size_t)s * 64))[lane];
  float* po = agg + (size_t)d * 64 + lane * 2;
  atomicAdd(po + 0, a.x * alpha);
  atomicAdd(po + 1, a.y * alpha);
}

// ----------------- fused: elu(agg2 + bias2) @ Wro + bro -> out ---------------
__global__ void readout(const float* __restrict__ agg2, const float* __restrict__ bias2,
                        const float* __restrict__ Wro, const float* __restrict__ bro,
                        float* __restrict__ out, int N) {
  const int n = (int)((blockIdx.x * blockDim.x + threadIdx.x) >> 5);
  const int lane = threadIdx.x & 31;
  if (n >= N) return;
  const float2 v = ((const float2*)(agg2 + (size_t)n * 64))[lane];
  const float2 b = ((const float2*)bias2)[lane];
  const float2 w = ((const float2*)Wro)[lane];
  float sum = elu(v.x + b.x) * w.x + elu(v.y + b.y) * w.y;
  sum += __shfl_xor(sum, 1);
  sum += __shfl_xor(sum, 2);
  sum += __shfl_xor(sum, 4);
  sum += __shfl_xor(sum, 8);
  sum += __shfl_xor(sum, 16);
  if (lane == 0) out[n] = sum + bro[0];
}

// ------------------------------------ host -----------------------------------
extern "C" void kernel_launch(void* const* d_in, const int* in_sizes, int n_in,
                              void* d_out, int out_size, void* d_ws, size_t ws_size,
                              hipStream_t stream) {
  (void)n_in; (void)out_size; (void)ws_size;
  const float* x     = (const float*)d_in[0];
  const int*   ei    = (const int*)d_in[1];
  const float* W1l   = (const float*)d_in[2];
  const float* b1l   = (const float*)d_in[3];
  const float* W1r   = (const float*)d_in[4];
  const float* b1r   = (const float*)d_in[5];
  const float* att1  = (const float*)d_in[6];
  const float* bias1 = (const float*)d_in[7];
  const float* W2l   = (const float*)d_in[8];
  const float* b2l   = (const float*)d_in[9];
  const float* W2r   = (const float*)d_in[10];
  const float* b2r   = (const float*)d_in[11];
  const float* att2  = (const float*)d_in[12];
  const float* bias2 = (const float*)d_in[13];
  const float* Wro   = (const float*)d_in[14];
  const float* bro   = (const float*)d_in[15];

  const int N = in_sizes[0] / 128;  // 50000 (multiple of 16)
  const int E = in_sizes[1] / 2;    // 1.6M
  const int* src = ei;
  const int* dst = ei + E;

  float* ws = (float*)d_ws;
  size_t off = 0;
  float* xl1 = ws + off; off += (size_t)N * 256;
  float* xr1 = ws + off; off += (size_t)N * 256;
  float* xl2 = ws + off; off += (size_t)N * 64;
  float* xr2 = ws + off; off += (size_t)N * 64;
  float* sc1 = ws + off; off += (size_t)E * 4;   // scores -> exp-weights (in place)
  float* sc2 = ws + off; off += (size_t)E;
  float* zbase = ws + off;                        // zero-initialized region
  float*    agg1 = ws + off; off += (size_t)N * 256;
  float*    den1 = ws + off; off += (size_t)N * 4;
  unsigned* mx1  = (unsigned*)(ws + off); off += (size_t)N * 4;
  float*    agg2 = ws + off; off += (size_t)N * 64;
  float*    den2 = ws + off; off += (size_t)N;
  unsigned* mx2  = (unsigned*)(ws + off); off += (size_t)N;
  const long zcount = (long)N * (256 + 4 + 4 + 64 + 1 + 1);

  // 1. zero accumulators / maxima (key 0 == -inf sentinel)
  {
    const long n4 = zcount / 4;
    zero_f4<<<(int)((n4 + 255) / 256), 256, 0, stream>>>((float4*)zbase, n4);
  }

  const int Mtiles = N / 16;              // 3125
  const int gx = (Mtiles + 7) / 8;        // 8 strip-tiles (waves) per workgroup
  const int ebl = (E + 7) / 8;            // 8 waves per 256-thread block, one wave/edge

  // 2. layer-1 node transforms (f32 WMMA, async-LDS staged B panel)
  gemm_wmma_lds<128><<<dim3(gx, 4), 256, 0, stream>>>(x, W1l, b1l, xl1, 256, Mtiles);
  gemm_wmma_lds<128><<<dim3(gx, 4), 256, 0, stream>>>(x, W1r, b1r, xr1, 256, Mtiles);

  // 3. layer-1 attention
  edge_score_l1<<<ebl, 256, 0, stream>>>(xl1, xr1, src, dst, att1, sc1, mx1, E);
  edge_softmax<4><<<(E * 4 + 255) / 256, 256, 0, stream>>>(dst, mx1, sc1, den1, E * 4);
  edge_aggr_l1<<<ebl, 256, 0, stream>>>(xl1, src, dst, sc1, den1, agg1, E);
  bias_elu<<<(N * 256 + 255) / 256, 256, 0, stream>>>(agg1, bias1, 255, N * 256);

  // 4. layer-2 node transforms (f32 WMMA), input = agg1 (h1)
  gemm_wmma_lds<256><<<dim3(gx, 1), 256, 0, stream>>>(agg1, W2l, b2l, xl2, 64, Mtiles);
  gemm_wmma_lds<256><<<dim3(gx, 1), 256, 0, stream>>>(agg1, W2r, b2r, xr2, 64, Mtiles);

  // 5. layer-2 attention
  edge_score_l2<<<ebl, 256, 0, stream>>>(xl2, xr2, src, dst, att2, sc2, mx2, E);
  edge_softmax<1><<<(E + 255) / 256, 256, 0, stream>>>(dst, mx2, sc2, den2, E);
  edge_aggr_l2<<<ebl, 256, 0, stream>>>(xl2, src, dst, sc2, den2, agg2, E);

  // 6. fused elu + readout
  readout<<<(N + 7) / 8, 256, 0, stream>>>(agg2, bias2, Wro, bro, (float*)d_out, N);
}